// CignDecisionLayer_40183714022063
// MI455X (gfx1250) — compile-verified
//
#include <hip/hip_runtime.h>
#include <cstdint>

#define B_SZ 32768
#define H_SZ 1024
#define N_SZ 4
#define C_SZ 1000
#define BN_EPS 1e-5f
#define IG_EPS 1e-30f
#define BALANCE 2.0f

#define CHUNKS 256          // stats pass: row chunks
#define ROWS_PER_CHUNK 128  // 256*128 = 32768
#define K3_BLOCKS 256       // main pass blocks
#define K3_ROWS 128         // rows per main-pass block (8 waves * 16 rows)
#define KC 64               // column chunk per wave in main pass

// workspace layout (float offsets)
#define OFF_SUM   0
#define OFF_SQ    (CHUNKS * H_SZ)
#define OFF_MEAN  (2 * CHUNKS * H_SZ)
#define OFF_RSTD  (OFF_MEAN + H_SZ)
#define OFF_WSUM  (OFF_RSTD + H_SZ)
#define OFF_PCN   (OFF_WSUM + 64)

typedef float v2f __attribute__((ext_vector_type(2)));
typedef float v8f __attribute__((ext_vector_type(8)));

// ---------------------------------------------------------------------------
// Kernel 1: per-chunk weighted column sums / sums-of-squares (streams h once)
// grid = (H/256, CHUNKS), block = 256
// ---------------------------------------------------------------------------
__global__ void k_colstats(const float* __restrict__ h, const int* __restrict__ mask,
                           float* __restrict__ ws) {
  int col  = blockIdx.x * 256 + threadIdx.x;
  int row0 = blockIdx.y * ROWS_PER_CHUNK;
  float s = 0.f, sq = 0.f;
  for (int r = 0; r < ROWS_PER_CHUNK; ++r) {
    int row  = row0 + r;
    float m  = (float)mask[row];
    float x  = h[(size_t)row * H_SZ + col];
    float mx = m * x;
    s  += mx;
    sq += mx * x;
    __builtin_prefetch(&h[(size_t)(row + 4) * H_SZ + col], 0, 0);
  }
  ws[OFF_SUM + blockIdx.y * H_SZ + col] = s;
  ws[OFF_SQ  + blockIdx.y * H_SZ + col] = sq;
}

// ---------------------------------------------------------------------------
// Kernel 2: finalize mean / rstd / wsum.  grid = H/256, block = 256
// ---------------------------------------------------------------------------
__global__ void k_finalize(const int* __restrict__ mask, float* __restrict__ ws) {
  __shared__ float red[256];
  int tid = threadIdx.x;
  float m = 0.f;
  for (int i = tid; i < B_SZ; i += 256) m += (float)mask[i];
  red[tid] = m;
  __syncthreads();
  for (int s = 128; s > 0; s >>= 1) {
    if (tid < s) red[tid] += red[tid + s];
    __syncthreads();
  }
  float wsum = red[0];

  int col = blockIdx.x * 256 + tid;
  float s = 0.f, sq = 0.f;
  for (int c = 0; c < CHUNKS; ++c) {
    s  += ws[OFF_SUM + c * H_SZ + col];
    sq += ws[OFF_SQ  + c * H_SZ + col];
  }
  float mean = s / wsum;
  float var  = sq / wsum - mean * mean;
  ws[OFF_MEAN + col] = mean;
  ws[OFF_RSTD + col] = rsqrtf(var + BN_EPS);
  if (blockIdx.x == 0 && tid == 0) ws[OFF_WSUM] = wsum;
}

// ---------------------------------------------------------------------------
// Kernel 3: fused normalize -> h_normed store -> WMMA fp32 GEMM (K=H, N=4)
//           -> softmax/argmax/routing -> per-block p_cn partials
// grid = K3_BLOCKS, block = 256 (8 waves, 16 rows each)
// ---------------------------------------------------------------------------
__global__ __launch_bounds__(256) void k_main(
    const float* __restrict__ h, const int* __restrict__ mask,
    const int* __restrict__ labels, const float* __restrict__ gamma,
    const float* __restrict__ beta, const float* __restrict__ W,
    const float* __restrict__ bvec, const float* __restrict__ ws,
    float* __restrict__ ws_pcn, float* __restrict__ out) {
  __shared__ float tile[8][16][68];          // padded: stride 68 floats -> conflict-free b64 reads
  __shared__ float ldsw[H_SZ * N_SZ];        // staged W (16 KB)
  __shared__ float lds_act[8][16][N_SZ];     // activations for row ops
  __shared__ float lds_pcn[C_SZ * N_SZ];     // block-local p_cn accumulator

  int tid = threadIdx.x;
  for (int i = tid; i < C_SZ * N_SZ; i += 256) lds_pcn[i] = 0.f;
  for (int i = tid; i < H_SZ * N_SZ; i += 256) ldsw[i] = W[i];
  __syncthreads();

  int wave = tid >> 5;
  int lane = tid & 31;
  int row0 = blockIdx.x * K3_ROWS + wave * 16;

  int  nIdx   = lane & 15;                  // B/C/D column this lane owns
  int  kOff   = (lane < 16) ? 0 : 2;        // A/B K sub-offset per half-wave
  bool nValid = (nIdx < N_SZ);
  float bn    = nValid ? bvec[nIdx] : 0.f;

  v8f acc = {0.f, 0.f, 0.f, 0.f, 0.f, 0.f, 0.f, 0.f};

  const float* wsMean = ws + OFF_MEAN;
  const float* wsRstd = ws + OFF_RSTD;

  for (int c0 = 0; c0 < H_SZ; c0 += KC) {
    int ca = c0 + lane, cb = ca + 32;
    float sa = gamma[ca] * wsRstd[ca];
    float sb = gamma[cb] * wsRstd[cb];
    float oa = beta[ca] - wsMean[ca] * sa;   // xn = x*s + o
    float ob = beta[cb] - wsMean[cb] * sb;
    for (int r = 0; r < 16; ++r) {
      size_t idx = (size_t)(row0 + r) * H_SZ + c0;
      float xa = h[idx + lane];
      float xb = h[idx + lane + 32];
      float na = xa * sa + oa;
      float nb = xb * sb + ob;
      out[idx + lane]      = na;             // h_normed, coalesced
      out[idx + lane + 32] = nb;
      tile[wave][r][lane]      = na;
      tile[wave][r][lane + 32] = nb;
      __builtin_prefetch(&h[idx + lane + KC], 0, 0);
    }
    __syncthreads();

    int arow = lane & 15;                    // A-matrix row this lane owns
    for (int k = 0; k < KC; k += 4) {
      v2f a = *reinterpret_cast<const v2f*>(&tile[wave][arow][k + kOff]); // K = k+kOff, k+kOff+1
      int kg = c0 + k + kOff;
      v2f bf;
      bf.x = nValid ? ldsw[kg * N_SZ + nIdx]       : 0.f;
      bf.y = nValid ? ldsw[(kg + 1) * N_SZ + nIdx] : 0.f;
      acc = __builtin_amdgcn_wmma_f32_16x16x4_f32(false, a, false, bf,
                                                  (short)0, acc, false, false);
    }
    __syncthreads();
  }

  // scatter accumulator (lane nIdx owns column nIdx; component i owns row i or i+8)
  if (nValid) {
    for (int i = 0; i < 8; ++i) {
      int M = i + ((lane < 16) ? 0 : 8);
      lds_act[wave][M][nIdx] = acc[i] + bn;
    }
  }
  __syncthreads();

  float wsum = ws[OFF_WSUM];
  if (lane < 16) {
    int row  = row0 + lane;
    float a0 = lds_act[wave][lane][0];
    float a1 = lds_act[wave][lane][1];
    float a2 = lds_act[wave][lane][2];
    float a3 = lds_act[wave][lane][3];

    int amx = 0; float best = a0;
    if (a1 > best) { best = a1; amx = 1; }
    if (a2 > best) { best = a2; amx = 2; }
    if (a3 > best) { best = a3; amx = 3; }

    int m = mask[row];
    size_t rbase = (size_t)B_SZ * H_SZ + 1 + (size_t)row * N_SZ;
    out[rbase + 0] = (m && amx == 0) ? 1.0f : 0.0f;
    out[rbase + 1] = (m && amx == 1) ? 1.0f : 0.0f;
    out[rbase + 2] = (m && amx == 2) ? 1.0f : 0.0f;
    out[rbase + 3] = (m && amx == 3) ? 1.0f : 0.0f;

    if (m) {
      float e0 = expf(a0 - best), e1 = expf(a1 - best);
      float e2 = expf(a2 - best), e3 = expf(a3 - best);
      float inv = (float)m / (wsum * (e0 + e1 + e2 + e3));
      int c = labels[row];
      atomicAdd(&lds_pcn[c * N_SZ + 0], inv * e0);
      atomicAdd(&lds_pcn[c * N_SZ + 1], inv * e1);
      atomicAdd(&lds_pcn[c * N_SZ + 2], inv * e2);
      atomicAdd(&lds_pcn[c * N_SZ + 3], inv * e3);
    }
  }
  __syncthreads();

  float* dst = ws_pcn + (size_t)blockIdx.x * (C_SZ * N_SZ);
  for (int i = tid; i < C_SZ * N_SZ; i += 256) dst[i] = lds_pcn[i];
}

// ---------------------------------------------------------------------------
// Kernel 4: reduce p_cn partials, compute info gain.  grid = 1, block = 1024
// ---------------------------------------------------------------------------
__global__ void k_ig(const float* __restrict__ ws_pcn, float* __restrict__ out) {
  __shared__ float pcn[C_SZ * N_SZ];
  __shared__ float pc[C_SZ];
  __shared__ float pn[N_SZ];
  __shared__ float red[1024];
  int tid = threadIdx.x;

  for (int i = tid; i < C_SZ * N_SZ; i += 1024) {
    float s = 0.f;
    for (int b = 0; b < K3_BLOCKS; ++b) s += ws_pcn[(size_t)b * (C_SZ * N_SZ) + i];
    pcn[i] = s;
  }
  __syncthreads();

  if (tid < C_SZ)
    pc[tid] = pcn[tid * 4] + pcn[tid * 4 + 1] + pcn[tid * 4 + 2] + pcn[tid * 4 + 3];
  if (tid < N_SZ) {
    float s = 0.f;
    for (int c = 0; c < C_SZ; ++c) s += pcn[c * 4 + tid];
    pn[tid] = s;
  }
  __syncthreads();

  float part = 0.f;
  for (int i = tid; i < C_SZ * N_SZ; i += 1024) {
    int c = i >> 2, n = i & 3;
    float v = pcn[i];
    part += v * (logf(v + IG_EPS) - logf(pc[c] + IG_EPS) - BALANCE * logf(pn[n] + IG_EPS));
  }
  red[tid] = part;
  __syncthreads();
  for (int s = 512; s > 0; s >>= 1) {
    if (tid < s) red[tid] += red[tid + s];
    __syncthreads();
  }
  if (tid == 0) out[(size_t)B_SZ * H_SZ] = -red[0];
}

// ---------------------------------------------------------------------------
extern "C" void kernel_launch(void* const* d_in, const int* in_sizes, int n_in,
                              void* d_out, int out_size, void* d_ws, size_t ws_size,
                              hipStream_t stream) {
  const int*   mask   = (const int*)d_in[0];
  const float* h      = (const float*)d_in[1];
  const int*   labels = (const int*)d_in[2];
  const float* gamma  = (const float*)d_in[3];
  const float* beta   = (const float*)d_in[4];
  const float* W      = (const float*)d_in[5];
  const float* bvec   = (const float*)d_in[6];
  float* out = (float*)d_out;
  float* ws  = (float*)d_ws;

  k_colstats<<<dim3(H_SZ / 256, CHUNKS), 256, 0, stream>>>(h, mask, ws);
  k_finalize<<<dim3(H_SZ / 256), 256, 0, stream>>>(mask, ws);
  k_main<<<dim3(K3_BLOCKS), 256, 0, stream>>>(h, mask, labels, gamma, beta, W, bvec,
                                              ws, ws + OFF_PCN, out);
  k_ig<<<1, 1024, 0, stream>>>(ws + OFF_PCN, out);
}